// CTCLoss_26044681683390
// MI455X (gfx1250) — compile-verified
//
#include <hip/hip_runtime.h>
#include <hip/hip_bf16.h>
#include <math.h>

// Problem constants (match reference)
#define TT 1000
#define BB 32
#define VV 1000
#define UU 100
#define SS 201              // 2U+1 extended states
#define EPAD 104            // blank + U labels, padded to 26*4 floats (B128-friendly)
#define NEG_EPS (-69.07755279f)   // log(1e-30)

// ---------------- CDNA5 async global->LDS helpers ----------------
typedef int v4i __attribute__((ext_vector_type(4)));
typedef __attribute__((address_space(1))) v4i gv4i;   // global-AS int4
typedef __attribute__((address_space(3))) v4i lv4i;   // LDS-AS int4

__device__ __forceinline__ void async_b128_to_lds(const float* g, float* l) {
#if __has_builtin(__builtin_amdgcn_global_load_async_to_lds_b128)
  __builtin_amdgcn_global_load_async_to_lds_b128((gv4i*)g, (lv4i*)l, 0, 0);
#else
  unsigned lds_off = (unsigned)(unsigned long long)l;
  asm volatile("global_load_async_to_lds_b128 %0, %1, off"
               :: "v"(lds_off), "v"(g) : "memory");
#endif
}

#if __has_builtin(__builtin_amdgcn_s_wait_asynccnt)
#define WAIT_ASYNC(n) __builtin_amdgcn_s_wait_asynccnt(n)
#else
#define WAIT_ASYNC(n) asm volatile("s_wait_asynccnt " #n ::: "memory")
#endif

// ---------------- Phase 1: log-softmax normalize + emission gather ----------------
// grid = T*B blocks, 128 threads (4 wave32). em layout: [t][b][EPAD],
// em[..][0] = logsoftmax blank, em[..][1+u] = logsoftmax of label targets[b,u].
// Row staged once via b128 loads; reductions are wave-level shuffles.
__global__ __launch_bounds__(128) void ctc_emissions(
    const float* __restrict__ lp, const int* __restrict__ targets,
    float* __restrict__ em) {
  __shared__ __align__(16) float row[VV];
  __shared__ float wred[4];
  const int tb   = blockIdx.x;       // t*B + b
  const int b    = tb % BB;
  const int tid  = threadIdx.x;
  const int lane = tid & 31;
  const int wv   = tid >> 5;

  // Stage row in LDS as float4 (V = 1000 = 250 * 4; row stride 4000B, 16B aligned)
  const float4* src4 = (const float4*)(lp + (size_t)tb * VV);
  float4* row4 = (float4*)row;
  float m = -INFINITY;
  for (int i = tid; i < 250; i += 128) {
    float4 v = src4[i];
    row4[i] = v;
    m = fmaxf(fmaxf(m, fmaxf(v.x, v.y)), fmaxf(v.z, v.w));
  }
  // wave32 max reduce, then cross-wave combine
  for (int off = 16; off; off >>= 1) m = fmaxf(m, __shfl_xor(m, off));
  if (lane == 0) wred[wv] = m;
  __syncthreads();
  m = fmaxf(fmaxf(wred[0], wred[1]), fmaxf(wred[2], wred[3]));

  // sum of exp (each thread re-reads only the elements it wrote)
  float ssum = 0.0f;
  for (int i = tid; i < 250; i += 128) {
    float4 v = row4[i];
    ssum += expf(v.x - m) + expf(v.y - m) + expf(v.z - m) + expf(v.w - m);
  }
  for (int off = 16; off; off >>= 1) ssum += __shfl_xor(ssum, off);
  __syncthreads();                 // everyone done reading wred (max phase)
  if (lane == 0) wred[wv] = ssum;
  __syncthreads();
  const float lse = m + logf(wred[0] + wred[1] + wred[2] + wred[3]);

  // Emission gather (row[] fully visible since before first barrier)
  float* dst = em + (size_t)tb * EPAD;
  for (int i = tid; i < EPAD; i += 128) {
    float e;
    if (i == 0)        e = row[0] - lse;
    else if (i <= UU)  e = row[targets[b * UU + (i - 1)]] - lse;
    else               e = NEG_EPS;  // pad, never consumed
    dst[i] = e;
  }
}

// ---------------- Phase 2: sequential alpha recursion ----------------
// grid = B blocks, 256 threads (8 wave32). One block per batch element.
// Emission row for step t+1 is async-DMA'd into a ping-pong LDS buffer while
// step t's 3-way logsumexp executes.
__global__ __launch_bounds__(256) void ctc_alpha(
    const float* __restrict__ em, const int* __restrict__ targets,
    const int* __restrict__ ilen, const int* __restrict__ tlen,
    float* __restrict__ out) {
  __shared__ float alpha[SS + 1];
  __shared__ __align__(16) float ebuf[2][EPAD];

  const int b   = blockIdx.x;
  const int tid = threadIdx.x;
  const int Tb  = ilen[b];                 // alpha frozen for t >= Tb
  const int s   = tid;
  const bool active = (s < SS);

  // Extended-label structure for this state
  int  eidx = 0;     // index into emission row: 0 = blank, 1+u = label u
  bool skip = false; // alpha[s-2] transition allowed
  if (active && (s & 1)) {
    eidx = 1 + (s >> 1);
    if (s >= 2) {
      // odd labels are always in [1,V) (non-blank); skip iff z[s] != z[s-2]
      skip = (targets[b * UU + (s >> 1)] != targets[b * UU + (s >> 1) - 1]);
    }
  }

  // alpha(t=0): s=0 -> blank emission, s=1 -> first label, else NEG_EPS
  if (active) {
    alpha[s] = (s < 2) ? em[(size_t)b * EPAD + eidx] : NEG_EPS;
  }

  // Prefetch emission row t=1 into ebuf[1] (wave 0, 26 lanes x 16B = 416B)
  if (tid < 26) {
    async_b128_to_lds(em + ((size_t)1 * BB + b) * EPAD + 4 * tid,
                      &ebuf[1][4 * tid]);
  }
  __syncthreads();

  for (int t = 1; t < Tb; ++t) {
    const int cur = t & 1, nxt = cur ^ 1;

    // Read previous alphas before any overwrite
    float a0 = -INFINITY, a1 = -INFINITY, a2 = -INFINITY;
    if (active) {
      a0 = alpha[s];
      if (s >= 1) a1 = alpha[s - 1];
      if (skip)   a2 = alpha[s - 2];
    }

    // Kick off DMA for step t+1 (clamped; tail issue is harmless)
    int tn = t + 1; if (tn >= Tb) tn = Tb - 1;
    if (tid < 26) {
      async_b128_to_lds(em + ((size_t)tn * BB + b) * EPAD + 4 * tid,
                        &ebuf[nxt][4 * tid]);
    }
    // Wave 0: ensure row t landed (in-order completion); leave t+1 in flight.
    WAIT_ASYNC(1);
    __syncthreads();   // ebuf[cur] visible to all waves; alpha reads done

    const float e  = ebuf[cur][eidx];
    const float mx = fmaxf(a0, fmaxf(a1, a2));
    const float nv = mx + logf(expf(a0 - mx) + expf(a1 - mx) + expf(a2 - mx)) + e;
    if (active) alpha[s] = nv;
    __syncthreads();   // writes visible before next step's reads
  }

  WAIT_ASYNC(0);
  if (tid == 0) {
    const int tl = tlen[b];
    const float x = alpha[2 * tl - 1];
    const float y = alpha[2 * tl];
    const float mx = fmaxf(x, y);
    out[b] = -(mx + logf(expf(x - mx) + expf(y - mx)));
  }
}

extern "C" void kernel_launch(void* const* d_in, const int* in_sizes, int n_in,
                              void* d_out, int out_size, void* d_ws, size_t ws_size,
                              hipStream_t stream) {
  const float* lp      = (const float*)d_in[0];  // [T,B,V] f32
  const int*   targets = (const int*)d_in[1];    // [B,U] i32
  const int*   in_len  = (const int*)d_in[2];    // [B] i32
  const int*   tg_len  = (const int*)d_in[3];    // [B] i32
  float* out = (float*)d_out;                    // [B] f32
  float* em  = (float*)d_ws;                     // [T,B,EPAD] f32 = ~13.3 MB

  ctc_emissions<<<TT * BB, 128, 0, stream>>>(lp, targets, em);
  ctc_alpha<<<BB, 256, 0, stream>>>(em, targets, in_len, tg_len, out);
}